// SpatialTemporalInteractiveGCN_10075993277108
// MI455X (gfx1250) — compile-verified
//
#include <hip/hip_runtime.h>

// ---------------------------------------------------------------------------
// SpatialTemporalInteractiveGCN fused kernel for MI455X (gfx1250, wave32).
//
// Reference math collapses (adj = row-normalized [I|I]) to, per (b,t,node):
//   y1 = x@W1^T + b1 ;  y2 = x@W2^T + b2            (D=64 GEMMs, fp32 WMMA)
//   a1 = c0*y1(t-1) + c1*y1(t)   (c0=adj[n,n], c1=adj[n,N+n]; y(t=-1)=bias)
//   a2 = c0*y2(t-1) + c1*y2(t)
//   z  = relu(a1*a2) + a1 + x ;  out = LayerNorm_D(z)*gamma + beta
//
// Memory-bound (~150MB traffic -> ~6.5us @ 23.3 TB/s), so we keep full fp32
// precision and use V_WMMA_F32_16X16X4_F32 for the GEMMs. One wave owns a
// 16-node x 64-feature tile; 256 wmma/wave. No LDS; no divergence (EXEC all 1s
// for WMMA). Cross-lane LayerNorm reductions use 16-lane shfl_xor (wave32).
// ---------------------------------------------------------------------------

typedef float v2f __attribute__((ext_vector_type(2)));
typedef float v8f __attribute__((ext_vector_type(8)));

constexpr int Bc = 8, Tc = 24, Nc = 1024, Dc = 64, WINc = 2;
constexpr float EPSc = 1e-5f;

__global__ __launch_bounds__(128)
void stgcn_fused_kernel(const float* __restrict__ x,
                        const float* __restrict__ W1, const float* __restrict__ b1,
                        const float* __restrict__ W2, const float* __restrict__ b2,
                        const float* __restrict__ gamma, const float* __restrict__ beta,
                        const float* __restrict__ adj,
                        float* __restrict__ out)
{
    const int lane = threadIdx.x & 31;
    const int wave = threadIdx.x >> 5;
    const int hi   = lane >> 4;      // 0: lanes 0-15, 1: lanes 16-31
    const int lo   = lane & 15;

    // block -> (b, t, group of 64 nodes); wave -> 16-node tile
    int blk = blockIdx.x;
    const int ng = blk % (Nc / 64); blk /= (Nc / 64);
    const int t  = blk % Tc;        blk /= Tc;
    const int b  = blk;
    const int nb = ng * 64 + wave * 16;          // first node of this wave's tile

    const float* xcur  = x + ((size_t)(b * Tc + t) * Nc + nb) * Dc;
    const float* xprev = xcur - (size_t)Nc * Dc; // only valid if t > 0
    const bool has_prev = (t > 0);               // wave-uniform -> EXEC stays full

    // Accumulators: 4 column chunks (e0 = 0,16,32,48), each a 16x16 f32 C tile.
    v8f a1c[4], a1p[4], a2c[4], a2p[4];
#pragma unroll
    for (int c = 0; c < 4; ++c) { a1c[c] = {}; a1p[c] = {}; a2c[c] = {}; a2p[c] = {}; }

    // A-fragment (f32 16x4): lane holds row (lane%16), K = 2*(lane/16)+{0,1}.
    const float* arow_c = xcur  + lo * Dc;
    const float* arow_p = xprev + lo * Dc;
    const v2f zero2 = {0.f, 0.f};

#pragma unroll
    for (int k0 = 0; k0 < Dc; k0 += 4) {
        const int koff = k0 + 2 * hi;
        v2f aC = *(const v2f*)(arow_c + koff);
        v2f aP = has_prev ? *(const v2f*)(arow_p + koff) : zero2;
#pragma unroll
        for (int c = 0; c < 4; ++c) {
            // B-fragment (f32 4x16): lane = output column e, K striped like A.
            // B[k,e] = W[e0+e, k0+k]  (out = X @ W^T)
            const int e = c * 16 + lo;
            v2f bW1 = *(const v2f*)(W1 + e * Dc + koff);
            v2f bW2 = *(const v2f*)(W2 + e * Dc + koff);
            a1c[c] = __builtin_amdgcn_wmma_f32_16x16x4_f32(false, aC, false, bW1,
                                                           (short)0, a1c[c], false, false);
            a1p[c] = __builtin_amdgcn_wmma_f32_16x16x4_f32(false, aP, false, bW1,
                                                           (short)0, a1p[c], false, false);
            a2c[c] = __builtin_amdgcn_wmma_f32_16x16x4_f32(false, aC, false, bW2,
                                                           (short)0, a2c[c], false, false);
            a2p[c] = __builtin_amdgcn_wmma_f32_16x16x4_f32(false, aP, false, bW2,
                                                           (short)0, a2p[c], false, false);
        }
    }

    // Per-node temporal-mix coefficients from the (block-diagonal) adjacency:
    //   c0 = adj[n, n]     (weight on t-1 window)
    //   c1 = adj[n, N + n] (weight on t   window)
    const int adjM = Nc * WINc;
    const int nrow = nb + lo;                    // node carried by this lane
    const float c0_mine = adj[(size_t)nrow * adjM + nrow];
    const float c1_mine = adj[(size_t)nrow * adjM + Nc + nrow];

    // ---- Elementwise epilogue in C-layout: lane holds column e = 16c + lo,
    // ---- VGPR v holds row m = v + 8*hi.  Stash z into a1c.
#pragma unroll
    for (int c = 0; c < 4; ++c) {
        const int e = c * 16 + lo;
        const float bb1 = b1[e];
        const float bb2 = b2[e];
#pragma unroll
        for (int v = 0; v < 8; ++v) {
            const int m = v + 8 * hi;
            // Broadcast row-m coefficients: row (v)     lives in lane v      (lo half),
            //                               row (8+v)   lives in lane 24+v   (hi half).
            const float c0 = __shfl(c0_mine, v + hi * 24, 32);
            const float c1 = __shfl(c1_mine, v + hi * 24, 32);
            const float y1  = a1c[c][v] + bb1;
            const float y1p = a1p[c][v] + bb1;   // t==0: acc==0 -> h(pad)=bias (matches ref)
            const float y2  = a2c[c][v] + bb2;
            const float y2p = a2p[c][v] + bb2;
            const float A1 = c0 * y1p + c1 * y1;
            const float A2 = c0 * y2p + c1 * y2;
            const float prod = A1 * A2;
            const float full = (prod > 0.f ? prod : 0.f) + A1;
            const float z = full + xcur[m * Dc + e];
            a1c[c][v] = z;
        }
    }

    // ---- LayerNorm over D=64: per row, reduce across 4 chunks x 16 lanes.
    float* outbase = out + ((size_t)(b * Tc + t) * Nc + nb) * Dc;
    float gv[4], bv[4];
#pragma unroll
    for (int c = 0; c < 4; ++c) { gv[c] = gamma[c * 16 + lo]; bv[c] = beta[c * 16 + lo]; }

#pragma unroll
    for (int v = 0; v < 8; ++v) {
        float s = a1c[0][v] + a1c[1][v] + a1c[2][v] + a1c[3][v];
        float q = a1c[0][v] * a1c[0][v] + a1c[1][v] * a1c[1][v]
                + a1c[2][v] * a1c[2][v] + a1c[3][v] * a1c[3][v];
#pragma unroll
        for (int msk = 1; msk <= 8; msk <<= 1) {   // stays within each 16-lane half
            s += __shfl_xor(s, msk, 32);
            q += __shfl_xor(q, msk, 32);
        }
        const float mu   = s * (1.0f / Dc);
        const float var  = q * (1.0f / Dc) - mu * mu;
        const float rstd = rsqrtf(var + EPSc);
        const int m = v + 8 * hi;
#pragma unroll
        for (int c = 0; c < 4; ++c) {
            const int e = c * 16 + lo;
            outbase[m * Dc + e] = (a1c[c][v] - mu) * rstd * gv[c] + bv[c];
        }
    }
}

extern "C" void kernel_launch(void* const* d_in, const int* in_sizes, int n_in,
                              void* d_out, int out_size, void* d_ws, size_t ws_size,
                              hipStream_t stream) {
    const float* x     = (const float*)d_in[0];
    const float* W1    = (const float*)d_in[1];
    const float* b1    = (const float*)d_in[2];
    const float* W2    = (const float*)d_in[3];
    const float* b2    = (const float*)d_in[4];
    const float* gamma = (const float*)d_in[5];
    const float* beta  = (const float*)d_in[6];
    const float* adj   = (const float*)d_in[7];
    float* out = (float*)d_out;

    // One wave per 16-node tile; 4 waves (128 threads) per block.
    dim3 block(128);
    dim3 grid(Bc * Tc * (Nc / 64));
    hipLaunchKernelGGL(stgcn_fused_kernel, grid, block, 0, stream,
                       x, W1, b1, W2, b2, gamma, beta, adj, out);
}